// Embedder_11458972746403
// MI455X (gfx1250) — compile-verified
//
#include <hip/hip_runtime.h>

typedef __attribute__((ext_vector_type(2))) float v2f;
typedef __attribute__((ext_vector_type(8))) float v8f;

#define DIN    128
#define DHID   64
#define DOUT   128
#define NGRAPHS 512

// ---------------------------------------------------------------- zero init
__global__ void gnn_zero_kernel(float* __restrict__ p, long long n) {
    long long i = (long long)blockIdx.x * blockDim.x + threadIdx.x;
    long long stride = (long long)gridDim.x * blockDim.x;
    for (; i < n; i += stride) p[i] = 0.0f;
}

// ------------------------------------------------- layer-1 edge scatter-add
// 32 lanes per edge, each lane moves 4 consecutive floats of the 128-wide row.
__global__ void gnn_scatter1_kernel(const float* __restrict__ x,
                                    const int*   __restrict__ esrc,
                                    const int*   __restrict__ edst,
                                    float* __restrict__ agg,
                                    float* __restrict__ deg,
                                    int n_edges) {
    int sub  = threadIdx.x >> 5;          // edge slot in block (8 per block)
    int lane = threadIdx.x & 31;
    int e = blockIdx.x * 8 + sub;
    if (e >= n_edges) return;
    int s = esrc[e];
    int d = edst[e];
    const float4 v = ((const float4*)(x + (long long)s * DIN))[lane];
    float* dst = agg + (long long)d * DIN + lane * 4;
    atomicAdd(dst + 0, v.x);
    atomicAdd(dst + 1, v.y);
    atomicAdd(dst + 2, v.z);
    atomicAdd(dst + 3, v.w);
    if (lane == 0) atomicAdd(deg + d, 1.0f);
}

// ------------------------------------------------- layer-2 edge scatter-add
// 16 lanes per edge over the 64-wide hidden row.
__global__ void gnn_scatter2_kernel(const float* __restrict__ h,
                                    const int*   __restrict__ esrc,
                                    const int*   __restrict__ edst,
                                    float* __restrict__ agg,
                                    int n_edges) {
    int sub  = threadIdx.x >> 4;          // 16 edges per 256-thread block
    int lane = threadIdx.x & 15;
    int e = blockIdx.x * 16 + sub;
    if (e >= n_edges) return;
    int s = esrc[e];
    int d = edst[e];
    const float4 v = ((const float4*)(h + (long long)s * DHID))[lane];
    float* dst = agg + (long long)d * DHID + lane * 4;
    atomicAdd(dst + 0, v.x);
    atomicAdd(dst + 1, v.y);
    atomicAdd(dst + 2, v.z);
    atomicAdd(dst + 3, v.w);
}

// ----------------------------------------------------------- GEMM1 (WMMA)
// h = relu( (agg/deg) @ W1_l + x @ W1_r + b1 ),   [16-node tile per wave]
__global__ void gnn_gemm1_kernel(const float* __restrict__ x,
                                 const float* __restrict__ agg,
                                 const float* __restrict__ deg,
                                 const float* __restrict__ Wl,   // [DIN,DHID]
                                 const float* __restrict__ Wr,   // [DIN,DHID]
                                 const float* __restrict__ bias, // [DHID]
                                 float* __restrict__ h,
                                 int n_tiles, int n_nodes) {
    int wave = threadIdx.x >> 5;
    int lane = threadIdx.x & 31;
    int tile = blockIdx.x * 4 + wave;
    if (tile >= n_tiles) return;                 // uniform per-wave exit

    int r16 = lane & 15;                         // A row / B col / C col
    int kb  = (lane >> 4) << 1;                  // K sub-offset: 0 or 2
    int node = tile * 16 + r16;
    int nclamp = node < n_nodes ? node : (n_nodes - 1);

    const float* xr = x   + (long long)nclamp * DIN;
    const float* ar = agg + (long long)nclamp * DIN;
    float inv = 1.0f / fmaxf(deg[nclamp], 1.0f);

    for (int j = 0; j < DHID / 16; ++j) {
        v8f c;
        float bv = bias[j * 16 + r16];
#pragma unroll
        for (int v = 0; v < 8; ++v) c[v] = bv;

        for (int k0 = 0; k0 < DIN; k0 += 4) {
            v2f am, ax, bl, br;
            am.x = ar[k0 + kb]     * inv;
            am.y = ar[k0 + kb + 1] * inv;
            ax.x = xr[k0 + kb];
            ax.y = xr[k0 + kb + 1];
            const float* wl = Wl + (long long)(k0 + kb) * DHID + j * 16 + r16;
            const float* wr = Wr + (long long)(k0 + kb) * DHID + j * 16 + r16;
            bl.x = wl[0]; bl.y = wl[DHID];
            br.x = wr[0]; br.y = wr[DHID];
            c = __builtin_amdgcn_wmma_f32_16x16x4_f32(false, am, false, bl,
                                                      (short)0, c, false, false);
            c = __builtin_amdgcn_wmma_f32_16x16x4_f32(false, ax, false, br,
                                                      (short)0, c, false, false);
        }
#pragma unroll
        for (int v = 0; v < 8; ++v) {
            int m = v + ((lane >> 4) << 3);      // C row
            int out_node = tile * 16 + m;
            if (out_node < n_nodes)
                h[(long long)out_node * DHID + j * 16 + r16] = fmaxf(c[v], 0.0f);
        }
    }
}

// ------------------------------------------- GEMM2 (WMMA) + fused graph pool
// node_out = (agg2/deg) @ W2_l + h @ W2_r + b2 ; psum[batch[n]] += node_out
__global__ void gnn_gemm2_pool_kernel(const float* __restrict__ h,
                                      const float* __restrict__ agg,
                                      const float* __restrict__ deg,
                                      const float* __restrict__ Wl,   // [DHID,DOUT]
                                      const float* __restrict__ Wr,   // [DHID,DOUT]
                                      const float* __restrict__ bias, // [DOUT]
                                      const int*   __restrict__ batch,
                                      float* __restrict__ psum,       // [NGRAPHS,DOUT]
                                      int n_tiles, int n_nodes) {
    int wave = threadIdx.x >> 5;
    int lane = threadIdx.x & 31;
    int tile = blockIdx.x * 4 + wave;
    if (tile >= n_tiles) return;

    int r16 = lane & 15;
    int kb  = (lane >> 4) << 1;
    int node = tile * 16 + r16;
    int nclamp = node < n_nodes ? node : (n_nodes - 1);

    const float* hr = h   + (long long)nclamp * DHID;
    const float* ar = agg + (long long)nclamp * DHID;
    float inv = 1.0f / fmaxf(deg[nclamp], 1.0f);

    for (int j = 0; j < DOUT / 16; ++j) {
        v8f c;
        float bv = bias[j * 16 + r16];
#pragma unroll
        for (int v = 0; v < 8; ++v) c[v] = bv;

        for (int k0 = 0; k0 < DHID; k0 += 4) {
            v2f am, ah, bl, br;
            am.x = ar[k0 + kb]     * inv;
            am.y = ar[k0 + kb + 1] * inv;
            ah.x = hr[k0 + kb];
            ah.y = hr[k0 + kb + 1];
            const float* wl = Wl + (long long)(k0 + kb) * DOUT + j * 16 + r16;
            const float* wr = Wr + (long long)(k0 + kb) * DOUT + j * 16 + r16;
            bl.x = wl[0]; bl.y = wl[DOUT];
            br.x = wr[0]; br.y = wr[DOUT];
            c = __builtin_amdgcn_wmma_f32_16x16x4_f32(false, am, false, bl,
                                                      (short)0, c, false, false);
            c = __builtin_amdgcn_wmma_f32_16x16x4_f32(false, ah, false, br,
                                                      (short)0, c, false, false);
        }
#pragma unroll
        for (int v = 0; v < 8; ++v) {
            int m = v + ((lane >> 4) << 3);
            int out_node = tile * 16 + m;
            if (out_node < n_nodes) {
                int g = batch[out_node];
                atomicAdd(&psum[(long long)g * DOUT + j * 16 + r16], c[v]);
            }
        }
    }
}

// ----------------------------------------------------------- per-graph count
__global__ void gnn_count_kernel(const int* __restrict__ batch,
                                 float* __restrict__ pcnt, int n_nodes) {
    int i = blockIdx.x * blockDim.x + threadIdx.x;
    if (i < n_nodes) atomicAdd(&pcnt[batch[i]], 1.0f);
}

// ------------------------------------------------------------------ finalize
__global__ void gnn_finalize_kernel(const float* __restrict__ psum,
                                    const float* __restrict__ pcnt,
                                    float* __restrict__ out, int n) {
    int i = blockIdx.x * blockDim.x + threadIdx.x;
    if (i < n) {
        int g = i / DOUT;
        out[i] = psum[i] / fmaxf(pcnt[g], 1.0f);
    }
}

// ------------------------------------------------------------------- launch
extern "C" void kernel_launch(void* const* d_in, const int* in_sizes, int n_in,
                              void* d_out, int out_size, void* d_ws, size_t ws_size,
                              hipStream_t stream) {
    const float* x    = (const float*)d_in[0];
    const float* W1l  = (const float*)d_in[1];
    const float* W1r  = (const float*)d_in[2];
    const float* b1   = (const float*)d_in[3];
    const float* W2l  = (const float*)d_in[4];
    const float* W2r  = (const float*)d_in[5];
    const float* b2   = (const float*)d_in[6];
    const int*   eidx = (const int*)d_in[7];   // [2, E] row-major
    const int*   batch= (const int*)d_in[8];

    const int n_nodes = in_sizes[0] / DIN;
    const int n_edges = in_sizes[7] / 2;
    const int* esrc = eidx;
    const int* edst = eidx + n_edges;

    // workspace carve-up (floats)
    float* ws = (float*)d_ws;
    long long off = 0;
    float* agg1 = ws + off; off += (long long)n_nodes * DIN;
    float* deg  = ws + off; off += n_nodes;
    float* h    = ws + off; off += (long long)n_nodes * DHID;
    float* agg2 = ws + off; off += (long long)n_nodes * DHID;
    float* psum = ws + off; off += (long long)NGRAPHS * DOUT;
    float* pcnt = ws + off; off += NGRAPHS;
    const long long total_ws = off;

    gnn_zero_kernel<<<2048, 256, 0, stream>>>(ws, total_ws);

    gnn_scatter1_kernel<<<(n_edges + 7) / 8, 256, 0, stream>>>(
        x, esrc, edst, agg1, deg, n_edges);

    const int n_tiles = (n_nodes + 15) / 16;
    gnn_gemm1_kernel<<<(n_tiles + 3) / 4, 128, 0, stream>>>(
        x, agg1, deg, W1l, W1r, b1, h, n_tiles, n_nodes);

    gnn_scatter2_kernel<<<(n_edges + 15) / 16, 256, 0, stream>>>(
        h, esrc, edst, agg2, n_edges);

    gnn_gemm2_pool_kernel<<<(n_tiles + 3) / 4, 128, 0, stream>>>(
        h, agg2, deg, W2l, W2r, b2, batch, psum, n_tiles, n_nodes);

    gnn_count_kernel<<<(n_nodes + 255) / 256, 256, 0, stream>>>(
        batch, pcnt, n_nodes);

    gnn_finalize_kernel<<<(NGRAPHS * DOUT + 255) / 256, 256, 0, stream>>>(
        psum, pcnt, (float*)d_out, NGRAPHS * DOUT);
}